// QPINN_1348619731650
// MI455X (gfx1250) — compile-verified
//
#include <hip/hip_runtime.h>
#include <stdint.h>

#define BLOCK 256

typedef __attribute__((ext_vector_type(4))) unsigned int u32x4;
typedef __attribute__((ext_vector_type(8))) unsigned int u32x8;

struct Cplx { float r, i; };

__device__ __forceinline__ uint32_t rfl(uint32_t v) {
    return (uint32_t)__builtin_amdgcn_readfirstlane((int)v);
}

// ---------------------------------------------------------------------------
// 1-D Tensor Data Mover load: n contiguous f32 from gptr -> LDS byte offset
// lds_off.  Descriptor layout per CDNA5 ISA:
//   group0: [1:0]=count=1, [63:32]=lds_addr, [120:64]=global_addr, [127:126]=2
//   group1: [17:16]=data_size(2 -> 4B), [79:48]=tensor_dim0, [111:80]=tensor_dim1,
//           [127:112]=tile_dim0, [143:128]=tile_dim1, [207:160]=tensor_dim0_stride
// Issued once per wave (EXEC ignored by TDM); tracked by TENSORcnt.
// ---------------------------------------------------------------------------
__device__ __forceinline__ void tdm_load_f32(uint32_t lds_off, const float* gptr,
                                             uint32_t n) {
    uint64_t ga = (uint64_t)(uintptr_t)gptr;
    u32x4 g0;
    g0.x = rfl(1u);                                  // count=1, user descriptor
    g0.y = rfl(lds_off);                             // lds_addr (bytes)
    g0.z = rfl((uint32_t)ga);                        // global_addr[31:0]
    g0.w = rfl(((uint32_t)(ga >> 32) & 0x01FFFFFFu)  // global_addr[56:32]
               | 0x80000000u);                       // type=2 ("image")
    u32x8 g1;
    g1.s0 = rfl(0x00020000u);                        // data_size=4B, wg_mask=0
    g1.s1 = rfl((n & 0xFFFFu) << 16);                // tensor_dim0[15:0]
    g1.s2 = rfl((n >> 16) | (1u << 16));             // tensor_dim0[31:16], tensor_dim1=1
    g1.s3 = rfl(n << 16);                            // tile_dim0 = n   (n <= 65535)
    g1.s4 = rfl(1u);                                 // tile_dim1 = 1
    g1.s5 = rfl(n);                                  // tensor_dim0_stride[31:0]
    g1.s6 = rfl(0u);
    g1.s7 = rfl(0u);
    asm volatile("tensor_load_to_lds %0, %1" :: "s"(g0), "s"(g1) : "memory");
}

// ---------------------------------------------------------------------------
// Setup kernel (runs once, 1 wave): precompute (cos(ph/2), sin(ph/2)) for all
// 14 QSP phases into ws, packed [phx1(3) | phx2(4) | pht1(3) | pht2(4)] pairs.
// ---------------------------------------------------------------------------
__global__ void phase_setup_kernel(const float* __restrict__ phx1,
                                   const float* __restrict__ phx2,
                                   const float* __restrict__ pht1,
                                   const float* __restrict__ pht2,
                                   float* __restrict__ ws) {
    const int k = (int)threadIdx.x;          // 0..13
    if (k < 14) {
        float ph;
        if      (k < 3)  ph = phx1[k];
        else if (k < 7)  ph = phx2[k - 3];
        else if (k < 10) ph = pht1[k - 7];
        else             ph = pht2[k - 10];
        const float h = 0.5f * ph;
        ws[2 * k]     = cosf(h);
        ws[2 * k + 1] = sinf(h);
    }
}

// ---------------------------------------------------------------------------
// <+| RZ(ph[L-1]) S(z) ... S(z) RZ(ph[0]) |+>  via 2-vector recursion, using
// precomputed phasor pairs pc[2k]=cos(ph_k/2), pc[2k+1]=sin(ph_k/2).
//   u <- S(z) u :  u0' = c*u0 + i*s*u1 ; u1' = i*s*u0 + c*u1
//   u <- RZ(p) u:  u0 *= e^{-ip/2} ; u1 *= e^{+ip/2}
// ---------------------------------------------------------------------------
template <int L>
__device__ __forceinline__ Cplx qsp_amp(const float* __restrict__ pc,
                                        float c, float sq) {
    Cplx u0, u1;
    u0.r = pc[0];  u0.i = -pc[1];
    u1.r = pc[0];  u1.i =  pc[1];

#pragma unroll
    for (int k = 1; k < L; ++k) {
        Cplx t0, t1;
        t0.r = c * u0.r - sq * u1.i;  t0.i = c * u0.i + sq * u1.r;
        t1.r = c * u1.r - sq * u0.i;  t1.i = c * u1.i + sq * u0.r;
        const float cs = pc[2 * k];
        const float sn = pc[2 * k + 1];
        u0.r = t0.r * cs + t0.i * sn;  u0.i = t0.i * cs - t0.r * sn; // * e^{-i ph/2}
        u1.r = t1.r * cs - t1.i * sn;  u1.i = t1.i * cs + t1.r * sn; // * e^{+i ph/2}
    }
    Cplx a;
    a.r = 0.5f * (u0.r + u1.r);
    a.i = 0.5f * (u0.i + u1.i);
    return a;
}

// ---------------------------------------------------------------------------
// Hadamard-test closed form:
//   out = Re( a_x * a_t ),  a_w = 0.5*(<+|QSP1_w|+> + <+|QSP2_w|+>)
// t/x tiles staged Global -> LDS by the Tensor Data Mover; phasors are
// uniform scalar loads from the precomputed cache.
// ---------------------------------------------------------------------------
__global__ void __launch_bounds__(BLOCK) qpinn_qsp_kernel(
    const float* __restrict__ tin, const float* __restrict__ xin,
    const float* __restrict__ pc,       // 28-float phasor cache in d_ws
    float* __restrict__ out, int n)
{
    extern __shared__ float sh[];            // [0,BLOCK)=t tile, [BLOCK,2*BLOCK)=x tile
    const int tid  = (int)threadIdx.x;
    const int base = (int)blockIdx.x * BLOCK;
    const int rem  = n - base;
    const uint32_t cnt = (uint32_t)(rem < BLOCK ? rem : BLOCK);

    if (threadIdx.x < 32u) {                 // wave 0 issues both DMA tiles
        tdm_load_f32(0u, tin + base, cnt);
        tdm_load_f32((uint32_t)(BLOCK * sizeof(float)), xin + base, cnt);
        __builtin_amdgcn_s_wait_tensorcnt(0);
    }
    __syncthreads();

    if (tid < (int)cnt) {
        const float tv = sh[tid];
        const float xv = sh[BLOCK + tid];

        float sx2 = 1.0f - xv * xv;  sx2 = sx2 > 0.0f ? sx2 : 0.0f;
        float st2 = 1.0f - tv * tv;  st2 = st2 > 0.0f ? st2 : 0.0f;
        const float sqx = sqrtf(sx2);
        const float sqt = sqrtf(st2);

        const Cplx qx1 = qsp_amp<3>(pc +  0, xv, sqx);
        const Cplx qx2 = qsp_amp<4>(pc +  6, xv, sqx);
        const Cplx qt1 = qsp_amp<3>(pc + 14, tv, sqt);
        const Cplx qt2 = qsp_amp<4>(pc + 20, tv, sqt);

        const float axr = 0.5f * (qx1.r + qx2.r);
        const float axi = 0.5f * (qx1.i + qx2.i);
        const float atr = 0.5f * (qt1.r + qt2.r);
        const float ati = 0.5f * (qt1.i + qt2.i);

        out[base + tid] = axr * atr - axi * ati;   // Re(a_x * a_t) = <Z_0>
    }
}

extern "C" void kernel_launch(void* const* d_in, const int* in_sizes, int n_in,
                              void* d_out, int out_size, void* d_ws, size_t ws_size,
                              hipStream_t stream) {
    const float* t    = (const float*)d_in[0];
    const float* x    = (const float*)d_in[1];
    const float* phx1 = (const float*)d_in[2];
    const float* phx2 = (const float*)d_in[3];
    const float* pht1 = (const float*)d_in[4];
    const float* pht2 = (const float*)d_in[5];
    float* out = (float*)d_out;
    float* pc  = (float*)d_ws;                          // 28 floats used

    const int n    = in_sizes[0];                       // 262144
    const int grid = (n + BLOCK - 1) / BLOCK;
    const size_t shmem = 2 * BLOCK * sizeof(float);

    phase_setup_kernel<<<1, 32, 0, stream>>>(phx1, phx2, pht1, pht2, pc);
    qpinn_qsp_kernel<<<grid, BLOCK, shmem, stream>>>(t, x, pc, out, n);
}